// DCNv4_56959856279789
// MI455X (gfx1250) — compile-verified
//
#include <hip/hip_runtime.h>
#include <hip/hip_bf16.h>

// ---------------------------------------------------------------------------
// DCNv4-style pipeline for gfx1250 (MI455X):
//   K1: fused offset(18ch)+mod(9ch) 3x3 conv  (VALU FMA, LDS-staged weights)
//   K2: deformable bilinear aggregation       (gather-bound, register-cached
//                                              corner idx/weights per pixel)
//   K3: 1x1 projection GEMM with V_WMMA_F32_16X16X4_F32, 32Mx64N per wave
// ---------------------------------------------------------------------------

#define Bsz   8
#define Cch   256
#define Hdim  64
#define Wdim  64
#define HW    (Hdim * Wdim)      // 4096
#define NOFF  18
#define NMOD  9
#define NOM   27                 // 18 + 9
#define COUT  256

typedef float v2f __attribute__((ext_vector_type(2)));
typedef float v8f __attribute__((ext_vector_type(8)));

#define WMMA_F32(A, Bf, Cf) \
    __builtin_amdgcn_wmma_f32_16x16x4_f32(false, (A), false, (Bf), (short)0, (Cf), false, false)

// ---------------------------------------------------------------------------
// Kernel 1: offset + modulator 3x3 conv, pad=1.
// One thread per output pixel; weights for one tap (27*256 floats = 27 KB)
// staged in LDS per iteration; uniform LDS reads broadcast across the wave.
// Output layout: offmod[b][o][hw], o in [0,27): 0..17 offset, 18..26 mod.
// ---------------------------------------------------------------------------
__global__ __launch_bounds__(256)
void k1_offmod_conv(const float* __restrict__ x,
                    const float* __restrict__ w_off,
                    const float* __restrict__ b_off,
                    const float* __restrict__ w_mod,
                    const float* __restrict__ b_mod,
                    float* __restrict__ offmod)
{
    __shared__ float lw[NOM * Cch];   // 27648 B

    const int tid = threadIdx.x;
    const int p   = blockIdx.x * 256 + tid;        // global pixel id
    const int b   = p >> 12;                       // / 4096
    const int yx  = p & (HW - 1);
    const int y   = yx >> 6;
    const int xc  = yx & 63;

    float acc[NOM];
#pragma unroll
    for (int o = 0; o < NOM; ++o)
        acc[o] = (o < NOFF) ? b_off[o] : b_mod[o - NOFF];

    const float* xb = x + (size_t)b * Cch * HW;

    for (int t9 = 0; t9 < 9; ++t9) {
        const int ky = t9 / 3, kx = t9 % 3;

        __syncthreads();
        // cooperative stage of this tap's weights: 6912 floats / 256 thr
        for (int i = tid; i < NOM * Cch; i += 256) {
            const int o = i >> 8;
            const int c = i & 255;
            lw[i] = (o < NOFF) ? w_off[(o * Cch + c) * 9 + t9]
                               : w_mod[((o - NOFF) * Cch + c) * 9 + t9];
        }
        __syncthreads();

        const int ys = y + ky - 1;
        const int xs = xc + kx - 1;
        if (ys < 0 || ys >= Hdim || xs < 0 || xs >= Wdim) continue;
        const int pix = ys * Wdim + xs;

        for (int c = 0; c < Cch; c += 4) {
            const float xv0 = xb[(c + 0) * HW + pix];
            const float xv1 = xb[(c + 1) * HW + pix];
            const float xv2 = xb[(c + 2) * HW + pix];
            const float xv3 = xb[(c + 3) * HW + pix];
#pragma unroll
            for (int o = 0; o < NOM; ++o) {
                const float4 w4 = *(const float4*)&lw[o * Cch + c];
                acc[o] += xv0 * w4.x + xv1 * w4.y + xv2 * w4.z + xv3 * w4.w;
            }
        }
    }

    float* om = offmod + (size_t)b * NOM * HW + yx;
#pragma unroll
    for (int o = 0; o < NOM; ++o)
        om[o * HW] = acc[o];
}

// ---------------------------------------------------------------------------
// Kernel 2: deformable aggregation.
// One thread per pixel. Precompute per-tap 4 clamped corner indices + 4
// corner weights (bilinear * mod * validity) into registers, then sweep all
// 256 channels: 36 gathers + 36 FMAs per channel. Adjacent lanes sample
// adjacent columns -> near-coalesced gathers; x is L2-resident (33 MB).
// ---------------------------------------------------------------------------
__global__ __launch_bounds__(256)
void k2_deform_agg(const float* __restrict__ x,
                   const float* __restrict__ offmod,
                   float* __restrict__ agg)
{
    const int p  = blockIdx.x * 256 + threadIdx.x;
    const int b  = p >> 12;
    const int yx = p & (HW - 1);
    const int y  = yx >> 6;
    const int xc = yx & 63;

    const float* om = offmod + (size_t)b * NOM * HW + yx;

    int   idx[9][4];
    float wgt[9][4];
#pragma unroll
    for (int k = 0; k < 9; ++k) {
        const float ox = om[(2 * k) * HW];       // comp 0 -> x offset
        const float oy = om[(2 * k + 1) * HW];   // comp 1 -> y offset
        const float m  = om[(NOFF + k) * HW];    // modulator
        const float sx = (float)xc + ox;
        const float sy = (float)y + oy;
        const float x0f = floorf(sx), y0f = floorf(sy);
        const float fx = sx - x0f, fy = sy - y0f;
        const int x0 = (int)x0f, y0 = (int)y0f;
#pragma unroll
        for (int cy = 0; cy < 2; ++cy) {
#pragma unroll
            for (int cx = 0; cx < 2; ++cx) {
                const int yi = y0 + cy, xi = x0 + cx;
                const bool v = (yi >= 0) && (yi < Hdim) && (xi >= 0) && (xi < Wdim);
                const int ycl = min(max(yi, 0), Hdim - 1);
                const int xcl = min(max(xi, 0), Wdim - 1);
                const float wb = (cy ? fy : 1.0f - fy) * (cx ? fx : 1.0f - fx) * m;
                idx[k][2 * cy + cx] = ycl * Wdim + xcl;
                wgt[k][2 * cy + cx] = v ? wb : 0.0f;
            }
        }
    }

    const float* xb = x + (size_t)b * Cch * HW;
    float* ab = agg + (size_t)b * Cch * HW + yx;

    for (int c = 0; c < Cch; ++c) {
        const float* xs = xb + c * HW;
        float a = 0.0f;
#pragma unroll
        for (int k = 0; k < 9; ++k) {
            a += wgt[k][0] * xs[idx[k][0]];
            a += wgt[k][1] * xs[idx[k][1]];
            a += wgt[k][2] * xs[idx[k][2]];
            a += wgt[k][3] * xs[idx[k][3]];
        }
        ab[c * HW] = a;
    }
}

// ---------------------------------------------------------------------------
// Kernel 3: 1x1 projection GEMM: out[b][m][n] = sum_k W[m][k]*agg[b][k][n]+bias
// Wave32 WMMA f32 16x16x4. Each wave owns a 32(M) x 64(N) tile: 2x4 = 8
// accumulator tiles. A fragments reused across 4 N-tiles, B fragments reused
// across 2 M-tiles -> ~1.25 VMEM ops per WMMA. K-loop: 64 steps of K=4.
//   A lane layout: rows M = lane&15; a.x/a.y = K = k0+2*(lane>>4) / +1
//   B lane layout: b.x/b.y = rows K = k0+2*(lane>>4) / +1; col N = tn+(lane&15)
//   C/D: VGPR j -> row tm + j + 8*(lane>>4), col tn + (lane&15)
// ---------------------------------------------------------------------------
__global__ __launch_bounds__(256)
void k3_proj_wmma(const float* __restrict__ agg,
                  const float* __restrict__ w_out,
                  const float* __restrict__ b_out,
                  float* __restrict__ out)
{
    const int wave = (blockIdx.x * 256 + threadIdx.x) >> 5;
    const int lane = threadIdx.x & 31;
    const int lo = lane & 15;
    const int hi = lane >> 4;

    // 512 wave-tiles per batch image: 8 M-supertiles(32) x 64 N-supertiles(64)
    const int b   = wave >> 9;
    const int rem = wave & 511;
    const int tm  = (rem >> 6) << 5;   // M supertile base (32 rows)
    const int tn  = (rem & 63) << 6;   // N supertile base (64 cols)

    const float* aggb = agg + (size_t)b * Cch * HW;

    v8f c00 = {}, c01 = {}, c02 = {}, c03 = {};
    v8f c10 = {}, c11 = {}, c12 = {}, c13 = {};
#pragma unroll
    for (int j = 0; j < 8; ++j) {
        const float bv0 = b_out[tm + j + 8 * hi];
        const float bv1 = b_out[tm + 16 + j + 8 * hi];
        c00[j] = bv0; c01[j] = bv0; c02[j] = bv0; c03[j] = bv0;
        c10[j] = bv1; c11[j] = bv1; c12[j] = bv1; c13[j] = bv1;
    }

    const float* arow0 = w_out + (size_t)(tm + lo) * Cch;        // M-tile 0
    const float* arow1 = arow0 + 16 * Cch;                       // M-tile 1

    for (int k0 = 0; k0 < Cch; k0 += 4) {
        const int ka = k0 + 2 * hi;            // even -> 8B aligned
        const v2f a0 = *(const v2f*)(arow0 + ka);
        const v2f a1 = *(const v2f*)(arow1 + ka);

        const float* bp = aggb + (size_t)ka * HW + tn + lo;
        v2f b0, b1, b2, b3;
        b0.x = bp[0];        b0.y = bp[HW];
        b1.x = bp[16];       b1.y = bp[HW + 16];
        b2.x = bp[32];       b2.y = bp[HW + 32];
        b3.x = bp[48];       b3.y = bp[HW + 48];

        c00 = WMMA_F32(a0, b0, c00);
        c01 = WMMA_F32(a0, b1, c01);
        c02 = WMMA_F32(a0, b2, c02);
        c03 = WMMA_F32(a0, b3, c03);
        c10 = WMMA_F32(a1, b0, c10);
        c11 = WMMA_F32(a1, b1, c11);
        c12 = WMMA_F32(a1, b2, c12);
        c13 = WMMA_F32(a1, b3, c13);
    }

    float* ob = out + (size_t)b * COUT * HW;
#pragma unroll
    for (int j = 0; j < 8; ++j) {
        const int m0 = tm + j + 8 * hi;
        const int m1 = m0 + 16;
        ob[m0 * HW + tn + lo +  0] = c00[j];
        ob[m0 * HW + tn + lo + 16] = c01[j];
        ob[m0 * HW + tn + lo + 32] = c02[j];
        ob[m0 * HW + tn + lo + 48] = c03[j];
        ob[m1 * HW + tn + lo +  0] = c10[j];
        ob[m1 * HW + tn + lo + 16] = c11[j];
        ob[m1 * HW + tn + lo + 32] = c12[j];
        ob[m1 * HW + tn + lo + 48] = c13[j];
    }
}

// ---------------------------------------------------------------------------
extern "C" void kernel_launch(void* const* d_in, const int* in_sizes, int n_in,
                              void* d_out, int out_size, void* d_ws, size_t ws_size,
                              hipStream_t stream) {
    (void)in_sizes; (void)n_in; (void)out_size; (void)ws_size;

    const float* x     = (const float*)d_in[0];
    const float* w_off = (const float*)d_in[1];
    const float* b_off = (const float*)d_in[2];
    const float* w_mod = (const float*)d_in[3];
    const float* b_mod = (const float*)d_in[4];
    const float* w_out = (const float*)d_in[5];
    const float* b_out = (const float*)d_in[6];
    float* out = (float*)d_out;

    // workspace carve-up
    const size_t offmod_bytes = (size_t)Bsz * NOM * HW * sizeof(float);   // 3.4 MB
    float* offmod = (float*)d_ws;
    float* agg    = (float*)((char*)d_ws + ((offmod_bytes + 255) & ~(size_t)255));

    const int npix = Bsz * HW;                 // 32768

    k1_offmod_conv<<<npix / 256, 256, 0, stream>>>(x, w_off, b_off, w_mod, b_mod, offmod);
    k2_deform_agg <<<npix / 256, 256, 0, stream>>>(x, offmod, agg);

    // 4096 wave-tiles total, 8 waves per 256-thread block -> 512 blocks
    k3_proj_wmma<<<(Bsz * 512) / 8, 256, 0, stream>>>(agg, w_out, b_out, out);
}